// ModelName_57294863728946
// MI455X (gfx1250) — compile-verified
//
#include <hip/hip_runtime.h>
#include <math.h>

typedef __attribute__((ext_vector_type(2))) float v2f;
typedef __attribute__((ext_vector_type(8))) float v8f;

#define HID 64
#define FIN 128
#define GNUM 128
#define NCLS 10

// ---- helpers -------------------------------------------------------------

// Order-preserving float<->int mapping for atomic max on floats.
__device__ __forceinline__ int f2ord(float f) {
  int i = __float_as_int(f);
  return (i >= 0) ? i : (i ^ 0x7fffffff);
}
__device__ __forceinline__ float ord2f(int i) {
  return __int_as_float((i >= 0) ? i : (i ^ 0x7fffffff));
}
// Native global_atomic_add_f32 (agent scope, relaxed).
__device__ __forceinline__ void atomAddF(float* p, float v) {
  __hip_atomic_fetch_add(p, v, __ATOMIC_RELAXED, __HIP_MEMORY_SCOPE_AGENT);
}

// ---- WMMA GEMM: C[M,64] = A[M,K] @ B[K,64], fp32 via V_WMMA_F32_16X16X4_F32
// One wave computes a 16x64 tile: A-fragment shared across the 4 column tiles.
// f32 16x4 A layout: lanes 0-15 hold M=0..15 with VGPR0=K+0,VGPR1=K+1;
// lanes 16-31 hold K+2,K+3 (per CDNA5 ISA 32-bit A 16x4 table).
//
// Out-of-range rows are handled by CLAMPING the A row pointer (an A row only
// feeds the same C row, and out-of-range C rows are not stored), so the
// K-loop has no per-lane predication -> no EXEC save/restore in the hot loop
// and the two A elements merge into one global_load_b64.
__global__ void gat_gemm_wmma(const float* __restrict__ A,
                              const float* __restrict__ B,
                              float* __restrict__ C, int M, int K) {
  int wave = (blockIdx.x * blockDim.x + threadIdx.x) >> 5;
  int lane = threadIdx.x & 31;
  int half = lane >> 4;          // 0: K+0/1, 1: K+2/3
  int l16  = lane & 15;
  int m0 = wave * 16;
  if (m0 >= M) return;           // wave-uniform: EXEC stays all-ones for WMMA

  v8f c0 = {}, c1 = {}, c2 = {}, c3 = {};
  int rowA = m0 + l16;
  if (rowA >= M) rowA = M - 1;   // clamp: contributes only to discarded rows
  const float* arow = A + (size_t)rowA * K + half * 2;

  for (int k = 0; k < K; k += 4) {
    v2f a;
    a.x = arow[k];
    a.y = arow[k + 1];
    const float* b0 = B + (size_t)(k + half * 2) * HID + l16;
    v2f b;
    b.x = b0[0];  b.y = b0[HID +  0];
    c0 = __builtin_amdgcn_wmma_f32_16x16x4_f32(false, a, false, b, (short)0, c0, false, false);
    b.x = b0[16]; b.y = b0[HID + 16];
    c1 = __builtin_amdgcn_wmma_f32_16x16x4_f32(false, a, false, b, (short)0, c1, false, false);
    b.x = b0[32]; b.y = b0[HID + 32];
    c2 = __builtin_amdgcn_wmma_f32_16x16x4_f32(false, a, false, b, (short)0, c2, false, false);
    b.x = b0[48]; b.y = b0[HID + 48];
    c3 = __builtin_amdgcn_wmma_f32_16x16x4_f32(false, a, false, b, (short)0, c3, false, false);
  }

  // C/D layout: VGPR j -> row m0 + half*8 + j, column = tile*16 + l16
  int rowBase = m0 + half * 8;
#pragma unroll
  for (int j = 0; j < 8; ++j) {
    int r = rowBase + j;
    if (r < M) {
      float* crow = C + (size_t)r * HID + l16;
      crow[0]  = c0[j];
      crow[16] = c1[j];
      crow[32] = c2[j];
      crow[48] = c3[j];
    }
  }
}

// ---- per-node attention coefficients: as[n]=h[n,:]·a_src, ad[n]=h[n,:]·a_dst
__global__ void node_alpha(const float* __restrict__ h,
                           const float* __restrict__ avs,
                           const float* __restrict__ avd,
                           float* __restrict__ out_s,
                           float* __restrict__ out_d, int Nn) {
  int wave = (blockIdx.x * blockDim.x + threadIdx.x) >> 5;
  int lane = threadIdx.x & 31;
  if (wave >= Nn) return;
  const float2 hv = *(const float2*)(h + (size_t)wave * HID + lane * 2);
  float ps = hv.x * avs[lane * 2] + hv.y * avs[lane * 2 + 1];
  float pd = hv.x * avd[lane * 2] + hv.y * avd[lane * 2 + 1];
#pragma unroll
  for (int off = 16; off; off >>= 1) {
    ps += __shfl_xor(ps, off, 32);
    pd += __shfl_xor(pd, off, 32);
  }
  if (lane == 0) { out_s[wave] = ps; out_d[wave] = pd; }
}

// ---- init kernels (d_ws is poisoned with 0xAA; must init every call) -----
__global__ void init_node(int* __restrict__ m, float* __restrict__ denom, int Nn) {
  int i = blockIdx.x * blockDim.x + threadIdx.x;
  if (i < Nn) { m[i] = (int)0x80000000; denom[i] = 0.0f; }
}
__global__ void zero_f32(float* __restrict__ p, int n) {
  int i = blockIdx.x * blockDim.x + threadIdx.x;
  if (i < n) p[i] = 0.0f;
}

// ---- edge pass 1: e = leaky_relu(as[src]+ad[dst], 0.2); segment max -------
__global__ void edge_pass1(const int* __restrict__ ei, int E, int Nn,
                           const float* __restrict__ asv, const float* __restrict__ adv,
                           float* __restrict__ ebuf, int* __restrict__ m) {
  int i = blockIdx.x * blockDim.x + threadIdx.x;
  int NE = E + Nn;
  if (i >= NE) return;
  int s, d;
  if (i < E) { s = ei[i]; d = ei[E + i]; } else { s = d = i - E; }
  float e = asv[s] + adv[d];
  e = (e > 0.0f) ? e : 0.2f * e;
  ebuf[i] = e;
  atomicMax(m + d, f2ord(e));
}

// ---- edge pass 2: ex = exp(e - m[dst]); segment sum of ex -----------------
__global__ void edge_pass2(const int* __restrict__ ei, int E, int Nn,
                           const int* __restrict__ m,
                           float* __restrict__ ebuf, float* __restrict__ denom) {
  int i = blockIdx.x * blockDim.x + threadIdx.x;
  int NE = E + Nn;
  if (i >= NE) return;
  int d = (i < E) ? ei[E + i] : (i - E);
  float ex = expf(ebuf[i] - ord2f(m[d]));
  ebuf[i] = ex;
  atomAddF(denom + d, ex);
}

// ---- edge pass 3: agg[dst,:] += alpha * h[src,:] (wave per edge) ----------
__global__ void edge_agg(const int* __restrict__ ei, int E, int Nn,
                         const float* __restrict__ h,
                         const float* __restrict__ ebuf,
                         const float* __restrict__ denom,
                         float* __restrict__ agg) {
  int wave = (blockIdx.x * blockDim.x + threadIdx.x) >> 5;
  int lane = threadIdx.x & 31;
  int NE = E + Nn;
  if (wave >= NE) return;
  int s, d;
  if (wave < E) { s = ei[wave]; d = ei[E + wave]; } else { s = d = wave - E; }
  float alpha = ebuf[wave] / denom[d];
  const float2 hv = *(const float2*)(h + (size_t)s * HID + lane * 2);
  float* o = agg + (size_t)d * HID + lane * 2;
  atomAddF(o,     hv.x * alpha);
  atomAddF(o + 1, hv.y * alpha);
}

// ---- relu(agg + bias) -----------------------------------------------------
__global__ void relu_bias(const float* __restrict__ agg, const float* __restrict__ b,
                          float* __restrict__ out, int n) {
  int i = blockIdx.x * blockDim.x + threadIdx.x;
  if (i >= n) return;
  float v = agg[i] + b[i & (HID - 1)];
  out[i] = v > 0.0f ? v : 0.0f;
}

// ---- global mean pool: pooled[g,:] += agg2[n,:]+b2 ; counts[g] += 1 -------
__global__ void pool_nodes(const float* __restrict__ agg2, const float* __restrict__ b2,
                           const int* __restrict__ batch,
                           float* __restrict__ pooled, float* __restrict__ counts, int Nn) {
  int wave = (blockIdx.x * blockDim.x + threadIdx.x) >> 5;
  int lane = threadIdx.x & 31;
  if (wave >= Nn) return;
  float2 v = *(const float2*)(agg2 + (size_t)wave * HID + lane * 2);
  v.x += b2[lane * 2];
  v.y += b2[lane * 2 + 1];
  int g = batch[wave];
  float* o = pooled + (size_t)g * HID + lane * 2;
  atomAddF(o,     v.x);
  atomAddF(o + 1, v.y);
  if (lane == 0) atomAddF(counts + g, 1.0f);
}

// ---- head: z = relu(mean @ lin_w + lin_b); logits = z @ cls_w + cls_b;
//      out = log_softmax(logits). One thread per graph (trivial FLOPs). -----
__global__ void head(const float* __restrict__ pooled, const float* __restrict__ counts,
                     const float* __restrict__ lin_w, const float* __restrict__ lin_b,
                     const float* __restrict__ cls_w, const float* __restrict__ cls_b,
                     float* __restrict__ out) {
  int g = threadIdx.x;
  if (g >= GNUM) return;
  float c = counts[g];
  c = c > 1.0f ? c : 1.0f;
  float p[HID];
#pragma unroll
  for (int k = 0; k < HID; ++k) p[k] = pooled[(size_t)g * HID + k] / c;
  float z[HID / 2];
  for (int j = 0; j < HID / 2; ++j) {
    float acc = lin_b[j];
    for (int k = 0; k < HID; ++k) acc += p[k] * lin_w[k * (HID / 2) + j];
    z[j] = acc > 0.0f ? acc : 0.0f;
  }
  float lg[NCLS];
  float mx = -3.0e38f;
  for (int j = 0; j < NCLS; ++j) {
    float acc = cls_b[j];
    for (int k = 0; k < HID / 2; ++k) acc += z[k] * cls_w[k * NCLS + j];
    lg[j] = acc;
    mx = acc > mx ? acc : mx;
  }
  float ssum = 0.0f;
  for (int j = 0; j < NCLS; ++j) ssum += expf(lg[j] - mx);
  float lse = mx + logf(ssum);
  for (int j = 0; j < NCLS; ++j) out[g * NCLS + j] = lg[j] - lse;
}

// ---- host side ------------------------------------------------------------
extern "C" void kernel_launch(void* const* d_in, const int* in_sizes, int n_in,
                              void* d_out, int out_size, void* d_ws, size_t ws_size,
                              hipStream_t stream) {
  const float* x     = (const float*)d_in[0];
  const int*   ei    = (const int*)d_in[1];
  /* d_in[2] edge_weight unused by GATConv (lin_edge is None) */
  const int*   batch = (const int*)d_in[3];
  const float* W1    = (const float*)d_in[4];
  const float* av_s1 = (const float*)d_in[5];
  const float* av_d1 = (const float*)d_in[6];
  const float* b1    = (const float*)d_in[7];
  const float* W2    = (const float*)d_in[8];
  const float* av_s2 = (const float*)d_in[9];
  const float* av_d2 = (const float*)d_in[10];
  const float* b2    = (const float*)d_in[11];
  const float* lin_w = (const float*)d_in[12];
  const float* lin_b = (const float*)d_in[13];
  const float* cls_w = (const float*)d_in[14];
  const float* cls_b = (const float*)d_in[15];

  const int Nn = in_sizes[0] / FIN;
  const int E  = in_sizes[1] / 2;
  const int NE = E + Nn;
  const int NH = Nn * HID;

  // workspace layout (floats)
  float* ws = (float*)d_ws;
  size_t off = 0;
  float* bufA  = ws + off; off += (size_t)NH;      // h1 -> h_in2 -> agg2
  float* bufB  = ws + off; off += (size_t)NH;      // agg1 -> h2
  float* asv   = ws + off; off += (size_t)Nn;
  float* adv   = ws + off; off += (size_t)Nn;
  float* denom = ws + off; off += (size_t)Nn;
  int*   mord  = (int*)(ws + off); off += (size_t)Nn;
  float* ebuf  = ws + off; off += (size_t)NE;
  float* pooled = ws + off; off += (size_t)GNUM * HID;
  float* counts = ws + off; off += (size_t)GNUM;   // contiguous with pooled

  const dim3 blk(256);
  const int nodeBlocks     = (Nn + 255) / 256;
  const int nodeWaveBlocks = (Nn + 7) / 8;
  const int edgeBlocks     = (NE + 255) / 256;
  const int edgeWaveBlocks = (NE + 7) / 8;
  const int nhBlocks       = (NH + 255) / 256;
  const int gemmBlocks     = (((Nn + 15) / 16) + 7) / 8;

  // ===== layer 1 =====
  gat_gemm_wmma<<<gemmBlocks, blk, 0, stream>>>(x, W1, bufA, Nn, FIN);
  node_alpha<<<nodeWaveBlocks, blk, 0, stream>>>(bufA, av_s1, av_d1, asv, adv, Nn);
  init_node<<<nodeBlocks, blk, 0, stream>>>(mord, denom, Nn);
  zero_f32<<<nhBlocks, blk, 0, stream>>>(bufB, NH);
  edge_pass1<<<edgeBlocks, blk, 0, stream>>>(ei, E, Nn, asv, adv, ebuf, mord);
  edge_pass2<<<edgeBlocks, blk, 0, stream>>>(ei, E, Nn, mord, ebuf, denom);
  edge_agg<<<edgeWaveBlocks, blk, 0, stream>>>(ei, E, Nn, bufA, ebuf, denom, bufB);
  relu_bias<<<nhBlocks, blk, 0, stream>>>(bufB, b1, bufA, NH);   // bufA = h_in2

  // ===== layer 2 =====
  gat_gemm_wmma<<<gemmBlocks, blk, 0, stream>>>(bufA, W2, bufB, Nn, HID); // bufB = h2
  node_alpha<<<nodeWaveBlocks, blk, 0, stream>>>(bufB, av_s2, av_d2, asv, adv, Nn);
  init_node<<<nodeBlocks, blk, 0, stream>>>(mord, denom, Nn);
  zero_f32<<<nhBlocks, blk, 0, stream>>>(bufA, NH);              // bufA = agg2
  edge_pass1<<<edgeBlocks, blk, 0, stream>>>(ei, E, Nn, asv, adv, ebuf, mord);
  edge_pass2<<<edgeBlocks, blk, 0, stream>>>(ei, E, Nn, mord, ebuf, denom);
  edge_agg<<<edgeWaveBlocks, blk, 0, stream>>>(ei, E, Nn, bufB, ebuf, denom, bufA);

  // ===== pooling + MLP head =====
  zero_f32<<<(GNUM * HID + GNUM + 255) / 256, blk, 0, stream>>>(pooled, GNUM * HID + GNUM);
  pool_nodes<<<nodeWaveBlocks, blk, 0, stream>>>(bufA, b2, batch, pooled, counts, Nn);
  head<<<1, 128, 0, stream>>>(pooled, counts, lin_w, lin_b, cls_w, cls_b, (float*)d_out);
}